// GraphAnchorSelector_8392366096620
// MI455X (gfx1250) — compile-verified
//
#include <hip/hip_runtime.h>

typedef __attribute__((ext_vector_type(2))) float v2f;
typedef __attribute__((ext_vector_type(8))) float v8f;
typedef __attribute__((ext_vector_type(4))) int v4i;

#define AS1 __attribute__((address_space(1)))
#define AS3 __attribute__((address_space(3)))

#define N_B 16
#define N_N 1024
#define N_P 256
#define N_D 64
#define K_TOP 26
#define P_TILE 8
#define NSUB 8
#define NPLANE 528  // 8p*64d + 16 dword pad -> 16-bank shift per n-plane
#define NSTEPS (N_N / NSUB)           // 128
#define GSTRIDE_ST (NSUB * N_P * N_D) // 131072 floats per sub-tile step
#define GSTRIDE_I  (2 * N_P * N_D)    // 32768 floats per chunk (2 n-planes)
#define LSTRIDE_I  (2 * NPLANE)       // 1056 floats per chunk in LDS

// workspace layout (float offsets)
#define WS_IMP   0
#define WS_SCORE 1024
#define WS_SUMS  (WS_SCORE + N_B * N_P)          // 5120
#define WS_ANCH  (WS_SUMS + N_B * N_P * N_D)     // 267264 (+16*1664 -> ~1.12MB)

#if __has_builtin(__builtin_amdgcn_global_load_async_to_lds_b128)
#define USE_ASYNC 1
#define GLD_ASYNC_B128(gp, lp) \
    __builtin_amdgcn_global_load_async_to_lds_b128((AS1 v4i*)(gp), (AS3 v4i*)(lp), 0, 0)
#endif

#if __has_builtin(__builtin_amdgcn_s_wait_asynccnt)
#define WAIT_ASYNC(n) __builtin_amdgcn_s_wait_asynccnt(n)
#else
#define WAIT_ASYNC(n) asm volatile("s_wait_asynccnt %0" ::"i"(n) : "memory")
#endif

// ---------------- Kernel 1: importance = column mean of adp ----------------
__global__ void k_importance(const float* __restrict__ adp, float* __restrict__ imp) {
    int col = blockIdx.x * 256 + threadIdx.x;  // 4 blocks x 256 = 1024 cols
    float s = 0.f;
#pragma unroll 8
    for (int i = 0; i < N_N; ++i) s += adp[i * N_N + col];
    imp[col] = s * (1.0f / N_N);
}

// ---------------- Kernel 2: fused mean-over-n (WMMA) + weighted-norm scores ----------------
__global__ void __launch_bounds__(256) k_main(const float* __restrict__ patches,
                                              const float* __restrict__ imp_g,
                                              float* __restrict__ score_ws,
                                              float* __restrict__ sums_ws) {
    __shared__ __align__(16) float tile[2][NSUB * NPLANE];  // 2 x 16.9 KB
    __shared__ float imp_lds[N_N];                          // 4 KB
    __shared__ float part_lds[32];                          // per-(wave,half,j2) score partials

    const int t    = threadIdx.x;
    const int lane = t & 31;
    const int w    = t >> 5;               // wave id 0..7 -> owns p = p0 + w
    const int p0   = blockIdx.x * P_TILE;  // p-tile base
    const int b    = blockIdx.y;

    for (int i = t; i < N_N; i += 256) imp_lds[i] = imp_g[i];

    const float* pb = patches + (size_t)b * N_N * N_P * N_D;

    // per-thread copy mapping: chunk i covers f = t + i*256
    const int frow = t >> 4, fd4 = t & 15;
    const int ns0 = frow >> 3, pp = frow & 7;
    const size_t gbase = ((size_t)ns0 * N_P + p0 + pp) * N_D + fd4 * 4;
    const int lbase = ns0 * NPLANE + pp * 64 + fd4 * 4;

    const v2f aones = {1.0f, 1.0f};  // all-ones A matrix: layout-invariant
    v8f acc[4];
#pragma unroll
    for (int dg = 0; dg < 4; ++dg) acc[dg] = 0.0f;
    float sacc0 = 0.f, sacc1 = 0.f;  // score partials (half-wave uniform, branch-free)

    // ---- shared compute body: one sub-tile (8 n-values) from LDS buffer tb ----
    auto computeTile = [&](const float* tb, int st) {
#pragma unroll
        for (int ks = 0; ks < 2; ++ks) {
            const int nloc = ks * 4;
            // WMMA: accumulate sum over 4 n for (p = p0+w, 16 d) per d-group
            {
                const int kh = lane >> 4;
                const int dl = lane & 15;
                const float* base0 = tb + (nloc + kh * 2 + 0) * NPLANE + w * 64;
                const float* base1 = tb + (nloc + kh * 2 + 1) * NPLANE + w * 64;
#pragma unroll
                for (int dg = 0; dg < 4; ++dg) {
                    v2f bv;
                    bv.x = base0[dg * 16 + dl];
                    bv.y = base1[dg * 16 + dl];
                    acc[dg] = __builtin_amdgcn_wmma_f32_16x16x4_f32(
                        false, aones, false, bv, (short)0, acc[dg], false, false);
                }
            }
            // sum-of-squares: 32 rows (4n x 8p) / 8 waves, half-wave per row
#pragma unroll
            for (int j2 = 0; j2 < 2; ++j2) {
                const int row  = w * 4 + j2 * 2 + (lane >> 4);
                const int noff = row >> 3, pr_unused = row & 7;
                (void)pr_unused;
                const float4 v =
                    *(const float4*)(tb + (nloc + noff) * NPLANE + pr_unused * 64 + (lane & 15) * 4);
                float s = v.x * v.x + v.y * v.y + v.z * v.z + v.w * v.w;
                s += __shfl_xor(s, 1, 32);
                s += __shfl_xor(s, 2, 32);
                s += __shfl_xor(s, 4, 32);
                s += __shfl_xor(s, 8, 32);
                // s and imp are uniform within each 16-lane half: accumulate on all lanes
                const int n = st * NSUB + nloc + noff;
                const float contrib = imp_lds[n] * sqrtf(s);
                if (j2 == 0) sacc0 += contrib; else sacc1 += contrib;
            }
        }
    };

#ifdef USE_ASYNC
    // ======== async-DMA pipeline: global -> LDS with no VGPR staging ========
#define ISSUE_ASYNC(st_, buf_)                                               \
    do {                                                                     \
        const float* g_ = pb + gbase + (size_t)(st_)*GSTRIDE_ST;             \
        float* l_ = &tile[(buf_)][lbase];                                    \
        GLD_ASYNC_B128(g_, l_);                                              \
        GLD_ASYNC_B128(g_ + 1 * GSTRIDE_I, l_ + 1 * LSTRIDE_I);              \
        GLD_ASYNC_B128(g_ + 2 * GSTRIDE_I, l_ + 2 * LSTRIDE_I);              \
        GLD_ASYNC_B128(g_ + 3 * GSTRIDE_I, l_ + 3 * LSTRIDE_I);              \
    } while (0)

    ISSUE_ASYNC(0, 0);
    for (int st = 0; st < NSTEPS; ++st) {
        const int buf = st & 1;
        if (st + 1 < NSTEPS) {
            ISSUE_ASYNC(st + 1, buf ^ 1);
            WAIT_ASYNC(4);  // newest 4 (tile st+1) may remain; tile st complete (in-order)
        } else {
            WAIT_ASYNC(0);
        }
        __syncthreads();  // all waves' DMA for tile st visible
        computeTile(tile[buf], st);
        __syncthreads();  // all reads of buf done before it is DMA-overwritten
    }
#else
    // ======== fallback: register-staged pipeline (named regs, no spill) ========
    float4 r0, r1, r2, r3;
#define LOADT(st_)                                                           \
    do {                                                                     \
        const float* g_ = pb + gbase + (size_t)(st_)*GSTRIDE_ST;             \
        r0 = *(const float4*)(g_);                                           \
        r1 = *(const float4*)(g_ + 1 * GSTRIDE_I);                           \
        r2 = *(const float4*)(g_ + 2 * GSTRIDE_I);                           \
        r3 = *(const float4*)(g_ + 3 * GSTRIDE_I);                           \
    } while (0)
#define STORET(buf_)                                                         \
    do {                                                                     \
        float* l_ = &tile[(buf_)][lbase];                                    \
        *(float4*)(l_) = r0;                                                 \
        *(float4*)(l_ + 1 * LSTRIDE_I) = r1;                                 \
        *(float4*)(l_ + 2 * LSTRIDE_I) = r2;                                 \
        *(float4*)(l_ + 3 * LSTRIDE_I) = r3;                                 \
    } while (0)

    LOADT(0);
    for (int st = 0; st < NSTEPS; ++st) {
        const int buf = st & 1;
        STORET(buf);
        if (st + 1 < NSTEPS) LOADT(st + 1);  // overlap with compute
        __syncthreads();
        computeTile(tile[buf], st);
    }
    __syncthreads();
#endif

    // ---- deterministic score combine: slot = w*4 + j2*2 + half; p = slot & 7 ----
    if ((lane & 15) == 0) {
        const int half = lane >> 4;
        part_lds[w * 4 + 0 + half] = sacc0;  // j2 = 0
        part_lds[w * 4 + 2 + half] = sacc1;  // j2 = 1
    }
    // extract mean sums: A = ones => all D rows identical; col = lane%16 holds d
    if (lane < 16) {
        float* dst = sums_ws + ((size_t)(b * N_P + p0 + w)) * N_D + lane;
#pragma unroll
        for (int dg = 0; dg < 4; ++dg) dst[dg * 16] = acc[dg][0];
    }
    __syncthreads();
    if (t < P_TILE)
        score_ws[b * N_P + p0 + t] =
            part_lds[t] + part_lds[t + 8] + part_lds[t + 16] + part_lds[t + 24];
}

// ---------------- Kernel 3: wave-parallel top-26 selection + gather means ----------------
__global__ void k_topk_gather(const float* __restrict__ score_ws,
                              const float* __restrict__ sums_ws,
                              float* __restrict__ anch_ws) {
    __shared__ int idxs[K_TOP];
    const int b = blockIdx.x, t = threadIdx.x;
    if (t < 32) {
        // lane t owns p = t*8 + i, scores held in registers
        float v[8];
        unsigned rm = 0;  // removed mask
#pragma unroll
        for (int i = 0; i < 8; ++i) v[i] = score_ws[b * N_P + t * 8 + i];
        for (int kk = 0; kk < K_TOP; ++kk) {
            // key = (score_bits << 32) | (256 - p): scores >= 0 so bits are order-monotone;
            // low field -> lowest index wins ties (matches jax.lax.top_k)
            unsigned long long best = 0ull;
#pragma unroll
            for (int i = 0; i < 8; ++i) {
                const int p = t * 8 + i;
                unsigned long long key =
                    ((rm >> i) & 1u)
                        ? 0ull
                        : ((unsigned long long)__float_as_uint(v[i]) << 32) |
                              (unsigned)(256 - p);
                best = key > best ? key : best;
            }
#pragma unroll
            for (int m = 16; m >= 1; m >>= 1) {
                unsigned long long o = __shfl_xor(best, m, 32);
                best = o > best ? o : best;
            }
            const int p = 256 - (int)(best & 0x1FFull);
            if (t == (p >> 3)) rm |= 1u << (p & 7);
            if (t == 0) idxs[kk] = p;
        }
    }
    __syncthreads();
    for (int q = t; q < K_TOP * N_D; q += 256) {
        const int kk = q >> 6, dd = q & 63;
        anch_ws[b * (K_TOP * N_D) + q] =
            sums_ws[((size_t)(b * N_P + idxs[kk])) * N_D + dd] * (1.0f / N_N);
    }
}

// ---------------- Kernel 4: broadcast anchors to [b*n, k, d] ----------------
__global__ void k_broadcast(const float* __restrict__ anch_ws, float4* __restrict__ out) {
    const unsigned kd4 = K_TOP * N_D / 4;  // 416 float4 per (b, j)
    const unsigned i = blockIdx.x * 256u + threadIdx.x;  // < 6,815,744 : fits u32
    const unsigned rowj = i / kd4;                       // flat (b*1024 + j)
    const unsigned q = i - rowj * kd4;
    const unsigned b = rowj >> 10;
    out[i] = ((const float4*)anch_ws)[b * kd4 + q];
}

extern "C" void kernel_launch(void* const* d_in, const int* in_sizes, int n_in,
                              void* d_out, int out_size, void* d_ws, size_t ws_size,
                              hipStream_t stream) {
    const float* patches = (const float*)d_in[0];
    const float* adp     = (const float*)d_in[1];
    float* ws    = (float*)d_ws;
    float* imp   = ws + WS_IMP;
    float* score = ws + WS_SCORE;
    float* sums  = ws + WS_SUMS;
    float* anch  = ws + WS_ANCH;

    hipLaunchKernelGGL(k_importance, dim3(4), dim3(256), 0, stream, adp, imp);
    hipLaunchKernelGGL(k_main, dim3(N_P / P_TILE, N_B), dim3(256), 0, stream,
                       patches, imp, score, sums);
    hipLaunchKernelGGL(k_topk_gather, dim3(N_B), dim3(256), 0, stream, score, sums, anch);
    // total output float4 = 16*1024*416 = 6,815,744 = 26624 * 256
    hipLaunchKernelGGL(k_broadcast, dim3(26624), dim3(256), 0, stream, anch, (float4*)d_out);
}